// PackedViterbi_47605417508874
// MI455X (gfx1250) — compile-verified
//
#include <hip/hip_runtime.h>

// Packed Viterbi forward (logsumexp semiring) for MI455X / gfx1250.
// theta: [T=256, B=32, S=128, S=128] fp32.  out: [B] fp32.
//
// Memory-bound: 512MB streamed once (~22us floor @ 23.3TB/s). One workgroup
// (8 wave32) per batch chain. Step rewritten in exp space so the j-reduction
// is an fp32 mat-vec via V_WMMA_F32_16X16X4_F32. theta tiles are staged to
// LDS with global_load_async_to_lds_b128 (ASYNCcnt) double buffering so
// HBM/L2 latency hides behind each step's WMMA chain; L2 warmed with
// global_prefetch_b8. Two WMMA accumulators break the serial C-chain.

typedef __attribute__((ext_vector_type(2))) float v2f;
typedef __attribute__((ext_vector_type(8))) float v8f;
typedef __attribute__((ext_vector_type(4))) int   v4i;

#define T_DIM 256
#define B_DIM 32
#define S_DIM 128
#define TILE  (S_DIM * S_DIM)   // 16384 floats = 64KB
#define PF_DIST 4

#if __has_builtin(__builtin_amdgcn_global_load_async_to_lds_b128) && \
    __has_builtin(__builtin_amdgcn_s_wait_asynccnt)
#define HAVE_ASYNC_LDS 1
#else
#define HAVE_ASYNC_LDS 0
#endif

typedef __attribute__((address_space(1))) v4i gv4i;   // global v4i
typedef __attribute__((address_space(3))) v4i lv4i;   // LDS v4i

// Stage one 64KB tile (16384 floats) global -> LDS with 256 threads.
// Each thread copies 16 chunks of 4 floats; lanes are 16B-contiguous so
// every instruction is a fully coalesced 512B/wave transfer.
__device__ __forceinline__ void stage_tile(const float* __restrict__ g,
                                           float* __restrict__ l, int tid) {
#if HAVE_ASYNC_LDS
#pragma unroll
  for (int c = 0; c < 16; ++c) {
    const int off = c * 1024 + tid * 4;
    __builtin_amdgcn_global_load_async_to_lds_b128(
        (gv4i*)(g + off), (lv4i*)(l + off), /*imm_off=*/0, /*cpol=*/0);
  }
#else
#pragma unroll
  for (int c = 0; c < 16; ++c) {
    const int off = c * 1024 + tid * 4;
    *(float4*)(l + off) = *(const float4*)(g + off);
  }
#endif
}

__device__ __forceinline__ void wait_stage() {
#if HAVE_ASYNC_LDS
  __builtin_amdgcn_s_wait_asynccnt(0);
#endif
}

__global__ __launch_bounds__(256, 1)
void PackedViterbi_kernel(const float* __restrict__ theta,
                          float* __restrict__ out) {
  const int b    = blockIdx.x;
  const int tid  = threadIdx.x;
  const int lane = tid & 31;
  const int wave = tid >> 5;          // 0..7 -> 16-row WMMA tile
  const int half = lane >> 4;         // 0: K={0,1}, 1: K={2,3} (A layout)
  const int r16  = lane & 15;
  const int koff = half * 2;
  const int i_row = wave * 16 + r16;

  extern __shared__ float smem[];
  // layout: [0,TILE) buf0 | [TILE,2*TILE) buf1 | sV[128] | sW[128] | sMax
  float* sV   = smem + 2 * TILE;
  float* sW   = sV + S_DIM;
  float* sMax = sW + S_DIM;

  if (tid < S_DIM) { sV[tid] = 0.0f; sW[tid] = 1.0f; }   // V0 = 0
  if (tid == 0) sMax[0] = 0.0f;

  const size_t t_stride = (size_t)B_DIM * TILE;
  const float* th_b = theta + (size_t)b * TILE;

  // Prologue: stage tile 0 into buf0.
  stage_tile(th_b, smem, tid);
  wait_stage();
  __syncthreads();

  float vmax = 0.0f;   // shift used to build current sW

  for (int t = 0; t < T_DIM; ++t) {
    // Warm L2 for tile t+PF_DIST (2 cachelines per thread).
    if (t + PF_DIST < T_DIM) {
      const float* pf = th_b + (size_t)(t + PF_DIST) * t_stride
                             + (size_t)tid * 64;
      __builtin_prefetch(pf, 0, 1);
      __builtin_prefetch(pf + 32, 0, 1);
    }
    // Kick off async staging of tile t+1 into the other buffer.
    if (t + 1 < T_DIM)
      stage_tile(th_b + (size_t)(t + 1) * t_stride,
                 smem + ((t + 1) & 1) * TILE, tid);

    // ---- s[i] = sum_j exp(theta[i,j]) * sW[j] via fp32 WMMA, K=128.
    //      Two accumulators (even/odd K-chunk) to break the serial
    //      WMMA->WMMA C-dependence chain.
    const float* M = smem + (t & 1) * TILE + i_row * S_DIM;
    v8f acc0 = {0.f, 0.f, 0.f, 0.f, 0.f, 0.f, 0.f, 0.f};
    v8f acc1 = {0.f, 0.f, 0.f, 0.f, 0.f, 0.f, 0.f, 0.f};
#pragma unroll
    for (int kb = 0; kb < 32; kb += 2) {
      const int j0 = kb * 4 + koff;
      const int j1 = j0 + 4;
      float2 m0 = *(const float2*)(M + j0);
      float2 w0 = *(const float2*)(sW + j0);
      float2 m1 = *(const float2*)(M + j1);
      float2 w1 = *(const float2*)(sW + j1);
      v2f a0, b0, a1, b1;
      a0.x = __expf(m0.x);  a0.y = __expf(m0.y);
      a1.x = __expf(m1.x);  a1.y = __expf(m1.y);
      b0.x = w0.x;          b0.y = w0.y;
      b1.x = w1.x;          b1.y = w1.y;
      acc0 = __builtin_amdgcn_wmma_f32_16x16x4_f32(
          false, a0, false, b0, (short)0, acc0, false, false);
      acc1 = __builtin_amdgcn_wmma_f32_16x16x4_f32(
          false, a1, false, b1, (short)0, acc1, false, false);
    }
    v8f acc = acc0 + acc1;

    // D layout: VGPR r = row M=r (lanes 0-15) / M=8+r (lanes 16-31).
    if (r16 == 0) {
#pragma unroll
      for (int r = 0; r < 8; ++r)
        sV[wave * 16 + half * 8 + r] = __logf(acc[r]) + vmax;
    }
    __syncthreads();

    // New shift: max over 128 states (wave 0), broadcast, rebuild sW.
    if (wave == 0) {
      float m = sV[lane];
      m = fmaxf(m, sV[lane + 32]);
      m = fmaxf(m, sV[lane + 64]);
      m = fmaxf(m, sV[lane + 96]);
#pragma unroll
      for (int off = 16; off > 0; off >>= 1)
        m = fmaxf(m, __shfl_xor(m, off, 32));
      if (lane == 0) sMax[0] = m;
    }
    __syncthreads();
    vmax = sMax[0];
    if (tid < S_DIM) sW[tid] = __expf(sV[tid] - vmax);

    // Staging of t+1 must be complete (own wave) before the barrier that
    // lets any wave read the other buffer next iteration.
    wait_stage();
    __syncthreads();
  }

  // Terminal: out[b] = vmax + log(sum_i exp(V[i]-vmax)).
  if (wave == 0) {
    float s = sW[lane] + sW[lane + 32] + sW[lane + 64] + sW[lane + 96];
#pragma unroll
    for (int off = 16; off > 0; off >>= 1)
      s += __shfl_xor(s, off, 32);
    if (lane == 0) out[b] = __logf(s) + vmax;
  }
}

extern "C" void kernel_launch(void* const* d_in, const int* in_sizes, int n_in,
                              void* d_out, int out_size, void* d_ws,
                              size_t ws_size, hipStream_t stream) {
  (void)in_sizes; (void)n_in; (void)d_ws; (void)ws_size; (void)out_size;
  const float* theta = (const float*)d_in[0];
  float* out = (float*)d_out;
  constexpr size_t smem_bytes = (size_t)(2 * TILE + 2 * S_DIM + 4)
                                * sizeof(float);
  PackedViterbi_kernel<<<B_DIM, 256, smem_bytes, stream>>>(theta, out);
}